// MAP_20023137534634
// MI455X (gfx1250) — compile-verified
//
#include <hip/hip_runtime.h>

typedef __attribute__((ext_vector_type(2))) float v2f;
typedef __attribute__((ext_vector_type(8))) float v8f;

// Problem shape is fixed by the reference; bake it in so the mean kernel's
// 1/M becomes an inline float constant (no v_div_* sequence) and all index
// math reduces to shifts/masks.
constexpr int kN = 32;    // batch
constexpr int kM = 16;    // reduced dim
constexpr int kP = 1024;  // rows
constexpr int kD = 512;   // feature dim
constexpr float kInvM = 1.0f / (float)kM;  // 0.0625f

// -------------------------------------------------------------------------
// Kernel 1: task_mean[n,p] = (1/M) * sum_m task[n,m,p]
//
// Cast as a matrix op: D = A x B + C with A = all (1/16)  (16x4 f32),
// B = task tile (4x16 f32, K=m-slice, N=p-tile). Four V_WMMA_F32_16X16X4_F32
// steps accumulate over m=0..15. Every row of D equals the p-tile's mean;
// since all rows of A are identical the sum is invariant to the exact
// K<->(lane-half,vgpr) slot mapping of B — only the column mapping
// (N = lane % 16) matters, which matches the documented striped layout.
// D row 0 lives in c-vgpr 0, lanes 0..15.
//
// One wave32 per 16-wide p-tile. Total tiles = N*P/16 = 2048.
// -------------------------------------------------------------------------
__global__ __launch_bounds__(256) void mean_wmma_kernel(
    const float* __restrict__ task, float* __restrict__ mean) {
  const int lane = threadIdx.x & 31;
  const int wave = threadIdx.x >> 5;
  const int tile = blockIdx.x * 8 + wave;      // 8 waves per 256-thread block
  constexpr int kTilesPerN = kP >> 4;          // 64
  const int n  = tile >> 6;                    // tile / 64
  const int p0 = (tile & (kTilesPerN - 1)) << 4;

  const int j     = lane & 15;  // column of B / p within tile
  const int khalf = lane >> 4;  // which half of K this lane holds

  v2f a; a.x = kInvM; a.y = kInvM;  // A = all 1/16
  v8f c = {};                        // accumulator

  const float* tbase = task + ((size_t)n * kM) * kP + (size_t)(p0 + j);

  // 4 WMMA steps (K=4 each) cover m = 0..15; EXEC is all-1s here.
#pragma unroll
  for (int m0 = 0; m0 < kM; m0 += 4) {
    v2f b;
    b.x = tbase[(size_t)(m0 + khalf * 2 + 0) * kP];
    b.y = tbase[(size_t)(m0 + khalf * 2 + 1) * kP];
    // (neg_a, A, neg_b, B, c_mod, C, reuse_a, reuse_b)
    c = __builtin_amdgcn_wmma_f32_16x16x4_f32(
        false, a, false, b, (short)0, c, false, false);
  }

  // Row 0 of D: vgpr 0, lanes 0..15 hold (M=0, N=lane).
  if (lane < 16) {
    mean[(size_t)n * kP + (size_t)(p0 + j)] = c[0];
  }
}

// -------------------------------------------------------------------------
// Kernel 2: out[n,p,d] = feat[n,p,d] * mean[n,p]
// Pure HBM stream: 128-bit loads/stores, one float4 per thread.
// mean index (idx >> 7) is uniform per wave -> scalar-cache/L1 broadcast.
// 128 MB of traffic -> ~5.6 us floor at 23.3 TB/s.
// -------------------------------------------------------------------------
__global__ __launch_bounds__(256) void scale_kernel(
    const float4* __restrict__ feat4, const float* __restrict__ mean,
    float4* __restrict__ out4, long total4) {
  long idx = (long)blockIdx.x * (long)blockDim.x + (long)threadIdx.x;
  if (idx < total4) {
    float  s = mean[idx >> 7];  // d/4 = 128 float4s per (n,p) row
    float4 f = feat4[idx];
    f.x *= s; f.y *= s; f.z *= s; f.w *= s;
    out4[idx] = f;
  }
}

extern "C" void kernel_launch(void* const* d_in, const int* in_sizes, int n_in,
                              void* d_out, int out_size, void* d_ws, size_t ws_size,
                              hipStream_t stream) {
  const float* task = (const float*)d_in[0];  // [32,16,1024] f32
  const float* feat = (const float*)d_in[1];  // [32,1024,512] f32
  float*       out  = (float*)d_out;          // [32,1024,512] f32
  float*       mean = (float*)d_ws;           // [32,1024] f32 scratch (128 KB)

  // Phase 1: mean over m via WMMA. 2048 waves = 256 blocks x 8 waves.
  constexpr int kTiles = (kN * kP) / 16;  // 2048
  mean_wmma_kernel<<<kTiles / 8, 256, 0, stream>>>(task, mean);

  // Phase 2: stream feat * mean. 4,194,304 float4s.
  const long total4 = (long)kN * (long)kP * (long)(kD / 4);
  scale_kernel<<<(int)(total4 / 256), 256, 0, stream>>>(
      (const float4*)feat, mean, (float4*)out, total4);
}